// DistanceLayer_78245714198970
// MI455X (gfx1250) — compile-verified
//
#include <hip/hip_runtime.h>

typedef __attribute__((ext_vector_type(2))) float v2f;
typedef __attribute__((ext_vector_type(8))) float v8f;

#define B_    32
#define T_    8192
#define C_    3
#define NS_   256
#define LS_   32
#define W_    8161          // T - LS + 1
#define KDIM  96            // LS * C
#define NKS   24            // KDIM / 4 WMMA k-steps
#define NTILE 511           // ceil(W / 16)

// ---------------- pre-kernel: sliding-window norms a2[b,w] and k2[s] ----------------
__global__ __launch_bounds__(256) void shapelet_pre_kernel(
    const float* __restrict__ data, const float* __restrict__ kern,
    float* __restrict__ a2, float* __restrict__ k2) {
  int idx = blockIdx.x * blockDim.x + threadIdx.x;
  const int nA = B_ * W_;
  if (idx < nA) {
    int b = idx / W_, w = idx - b * W_;
    const float* p = data + b * (T_ * C_) + w * C_;   // 96 contiguous floats
    float s = 0.f;
#pragma unroll 8
    for (int j = 0; j < KDIM; ++j) { float x = p[j]; s += x * x; }
    a2[idx] = s;
  } else if (idx < nA + NS_) {
    int sId = idx - nA;
    const float* p = kern + sId * KDIM;
    float s = 0.f;
#pragma unroll 8
    for (int j = 0; j < KDIM; ++j) { float x = p[j]; s += x * x; }
    k2[sId] = s;
  }
}

// ---------------- main kernel: WMMA cross-correlation + min reduction ----------------
// grid = (NS/16, B), block = 256 (8 waves). Wave handles 16 shapelets x 16 windows tiles.
__global__ __launch_bounds__(256) void shapelet_wmma_kernel(
    const float* __restrict__ data, const float* __restrict__ kern,
    const float* __restrict__ a2, const float* __restrict__ k2g,
    float* __restrict__ out) {
  const int tid   = threadIdx.x;
  const int lane  = tid & 31;
  const int wid   = tid >> 5;
  const int b     = blockIdx.y;
  const int s0    = blockIdx.x << 4;
  const int n     = lane & 15;      // N (shapelet) / M (window) sub-index
  const int khalf = lane >> 4;      // which K pair of the 16x16x4 fragment

  // Preload all B fragments (kernel tile) for K = 0..95 : 24 x v2f = 48 VGPRs, reused
  // across every window tile. B[k, n] = kern[(s0+n)*96 + k]; lane holds k = 4*kk + 2*khalf + {0,1}.
  const float* kb = kern + (s0 + n) * KDIM + (khalf << 1);
  v2f Bv[NKS];
#pragma unroll
  for (int kk = 0; kk < NKS; ++kk) {
    v2f t; t.x = kb[kk * 4]; t.y = kb[kk * 4 + 1];
    Bv[kk] = t;
  }
  const float  k2n = k2g[s0 + n];
  const float* db  = data + b * (T_ * C_);
  const float* a2b = a2 + b * W_;

  float minv = 3.402823466e+38f;

  for (int wt = wid; wt < NTILE; wt += 8) {           // wave-uniform: EXEC stays all-ones
    const int w0 = wt << 4;
    int wa = w0 + n; if (wa > W_ - 1) wa = W_ - 1;    // clamp: duplicate last window (min-safe)
    const float* ap = db + wa * C_ + (khalf << 1);    // A[m,k] = data_b[m_window*3 + k]

    v8f acc = {};
#pragma unroll
    for (int kk = 0; kk < NKS; ++kk) {
      v2f a; a.x = ap[kk * 4]; a.y = ap[kk * 4 + 1];
      acc = __builtin_amdgcn_wmma_f32_16x16x4_f32(
          /*neg_a=*/false, a, /*neg_b=*/false, Bv[kk],
          /*c_mod=*/(short)0, acc, /*reuse_a=*/false, /*reuse_b=*/false);
    }

    // D[m = r + 8*khalf, n]; dist = ||a||^2 + ||k||^2 - 2 a.k, fold into running min
#pragma unroll
    for (int r = 0; r < 8; ++r) {
      int w = w0 + r + (khalf << 3);
      if (w > W_ - 1) w = W_ - 1;
      float dist = a2b[w] + k2n - 2.0f * acc[r];
      minv = fminf(minv, dist);
    }
  }

  // Block reduction: red[wid*32 + khalf*16 + n]; tree with stride>=16 combines same n.
  __shared__ float red[256];
  red[tid] = minv;
  __syncthreads();
#pragma unroll
  for (int off = 128; off >= 16; off >>= 1) {
    if (tid < off) red[tid] = fminf(red[tid], red[tid + off]);
    __syncthreads();
  }
  if (tid < 16) out[b * NS_ + s0 + tid] = red[tid] * (1.0f / (float)LS_);
}

extern "C" void kernel_launch(void* const* d_in, const int* in_sizes, int n_in,
                              void* d_out, int out_size, void* d_ws, size_t ws_size,
                              hipStream_t stream) {
  (void)in_sizes; (void)n_in; (void)out_size; (void)ws_size;
  const float* data = (const float*)d_in[0];   // [32, 8192, 3]
  const float* kern = (const float*)d_in[1];   // [256, 32, 3]
  float* out = (float*)d_out;                  // [32, 256]

  float* a2 = (float*)d_ws;                    // [32, 8161]  (~1.04 MB)
  float* k2 = a2 + B_ * W_;                    // [256]

  const int totalPre = B_ * W_ + NS_;
  shapelet_pre_kernel<<<(totalPre + 255) / 256, 256, 0, stream>>>(data, kern, a2, k2);

  dim3 grid(NS_ / 16, B_);                     // (16, 32) = 512 blocks x 8 waves
  shapelet_wmma_kernel<<<grid, 256, 0, stream>>>(data, kern, a2, k2, out);
}